// Equiformer_16192026706331
// MI455X (gfx1250) — compile-verified
//
#include <hip/hip_runtime.h>

typedef __attribute__((ext_vector_type(2))) float v2f;
typedef __attribute__((ext_vector_type(4))) float v4f;
typedef __attribute__((ext_vector_type(8))) float v8f;

#define NHEAD 8
#define NPOS  131072          // I*J = 4096*32
#define TILE  16
#define SOUT  48
#define VOUT  16

// ---- LDS layout (in floats), bank-conflict-free padded strides ----
// weights, permuted [c][h][pos][m] with pos-stride 18 (18*k mod 64 distinct for k=0..15)
#define W_BASE      0
#define W_POSSTRIDE 18
#define W_HSTRIDE   (16 * W_POSSTRIDE)       // 288
#define W_CSTRIDE   (8 * W_HSTRIDE)          // 2304
#define W_SIZE      (4 * W_CSTRIDE)          // 9216
// v, [h][pos][m*3+k] with pos-stride 50 (50*k mod 64 distinct for k=0..15)
#define V_BASE      W_SIZE
#define V_POSSTRIDE 50
#define V_HSTRIDE   (16 * V_POSSTRIDE)       // 800
#define V_SIZE      (8 * V_HSTRIDE)          // 6400
#define LDS_FLOATS  (W_SIZE + V_SIZE)        // 15616 floats = 62464 B (< 64 KB)
// output staging (after barrier, reuses [0, 12288) ): per head 1536 floats

__global__ __launch_bounds__(256) void equi_wmma_kernel(
    const float* __restrict__ s_g,     // (H, NPOS, 16)
    const float* __restrict__ v_g,     // (H, NPOS, 16, 3)
    const float* __restrict__ r_g,     // (NPOS, 3)
    const float* __restrict__ w_g,     // (NPOS, 4, 8, 16)
    const float* __restrict__ ws_g,    // (H, 48, 32)
    const float* __restrict__ bs_g,    // (H, 48)
    const float* __restrict__ wv_g,    // (H, 16, 32)
    float* __restrict__ sout_g,        // (H, NPOS, 48)
    float* __restrict__ vout_g)        // (H, NPOS, 16, 3)
{
    __shared__ float lds[LDS_FLOATS];
    const int tid = threadIdx.x;
    const int p0  = blockIdx.x * TILE;

    // ---------- stage weights: 16 pos x 512 f32 contiguous -> LDS permuted ----------
    {
        const v4f* wg4 = (const v4f*)(w_g + (size_t)p0 * 512);
        #pragma unroll
        for (int i = 0; i < 8; ++i) {
            int g4 = tid + i * 256;                 // 0..2047 float4s
            v4f val = wg4[g4];
            int m4  = g4 & 3;
            int h   = (g4 >> 2) & 7;
            int c   = (g4 >> 5) & 3;
            int pos = g4 >> 7;
            int a = W_BASE + c * W_CSTRIDE + h * W_HSTRIDE + pos * W_POSSTRIDE + m4 * 4;
            *(v2f*)&lds[a]     = val.xy;            // pos*18 keeps only 8B alignment
            *(v2f*)&lds[a + 2] = val.zw;
        }
    }
    // ---------- stage v: per head 16 pos x 48 f32 contiguous -> LDS padded ----------
    {
        #pragma unroll
        for (int i = 0; i < 6; ++i) {
            int g4 = tid + i * 256;                 // 0..1535 float4s
            int h   = g4 / 192;
            int rem = g4 - h * 192;
            int pos = rem / 12;
            int q   = rem - pos * 12;
            const v4f* vg4 = (const v4f*)(v_g + ((size_t)h * NPOS + p0) * 48);
            v4f val = vg4[pos * 12 + q];
            int a = V_BASE + h * V_HSTRIDE + pos * V_POSSTRIDE + q * 4;
            *(v2f*)&lds[a]     = val.xy;
            *(v2f*)&lds[a + 2] = val.zw;
        }
    }
    __syncthreads();

    // ---------- per-wave compute: wave w == head h, 16 positions ----------
    const int lane = tid & 31;
    const int h    = tid >> 5;
    const int half = lane >> 4;
    const int pos  = lane & 15;
    const int p    = p0 + pos;

    const float r0 = r_g[p * 3 + 0];
    const float r1 = r_g[p * 3 + 1];
    const float r2 = r_g[p * 3 + 2];

    v8f acc_s0 = {}, acc_s1 = {}, acc_s2 = {};
    v8f acc_v0 = {}, acc_v1 = {}, acc_v2 = {};

    const float* wsh = ws_g + (size_t)h * (SOUT * 32);
    const float* wvh = wv_g + (size_t)h * (VOUT * 32);
    const float* sph = s_g + ((size_t)h * NPOS + p) * 16;
    const float* wlp = &lds[W_BASE + h * W_HSTRIDE + pos * W_POSSTRIDE];   // + c*W_CSTRIDE + m
    const float* vlp = &lds[V_BASE + h * V_HSTRIDE + pos * V_POSSTRIDE];   // + m*3 + k

    #pragma unroll
    for (int t = 0; t < 8; ++t) {
        const int c0 = 4 * t + 2 * half;    // K columns (c0, c0+1) for this lane
        const int m0 = c0 & 15;

        // A fragments (16x4 f32): lane holds row (16*mi+pos), cols c0,c0+1
        v2f a0 = *(const v2f*)&wsh[(0 * 16 + pos) * 32 + c0];
        v2f a1 = *(const v2f*)&wsh[(1 * 16 + pos) * 32 + c0];
        v2f a2 = *(const v2f*)&wsh[(2 * 16 + pos) * 32 + c0];
        v2f av = *(const v2f*)&wvh[pos * 32 + c0];

        // B fragments (4x16 f32): lane holds col pos, rows c0,c0+1
        v2f bs, bv0, bv1, bv2;
        if (t < 4) {
            // channels 0..15: ss = w_ss*s ; sv = (w_sv*s)*r
            v2f sv  = *(const v2f*)&sph[m0];
            v2f wss = *(const v2f*)&wlp[0 * W_CSTRIDE + m0];
            v2f wsv = *(const v2f*)&wlp[1 * W_CSTRIDE + m0];
            bs = wss * sv;
            v2f q = wsv * sv;
            bv0 = q * r0; bv1 = q * r1; bv2 = q * r2;
        } else {
            // channels 16..31: vv = w_vv*(v.r) ; vs = w_vs*v
            v2f wvs = *(const v2f*)&wlp[2 * W_CSTRIDE + m0];
            v2f wvv = *(const v2f*)&wlp[3 * W_CSTRIDE + m0];
            float va0 = vlp[m0 * 3 + 0], va1 = vlp[m0 * 3 + 1], va2 = vlp[m0 * 3 + 2];
            float vb0 = vlp[m0 * 3 + 3], vb1 = vlp[m0 * 3 + 4], vb2 = vlp[m0 * 3 + 5];
            float da = va0 * r0 + va1 * r1 + va2 * r2;
            float db = vb0 * r0 + vb1 * r1 + vb2 * r2;
            bs.x  = wvv.x * da;  bs.y  = wvv.y * db;
            bv0.x = wvs.x * va0; bv0.y = wvs.y * vb0;
            bv1.x = wvs.x * va1; bv1.y = wvs.y * vb1;
            bv2.x = wvs.x * va2; bv2.y = wvs.y * vb2;
        }

        // D(16x16,f32) += A(16x4,f32) x B(4x16,f32)
        acc_s0 = __builtin_amdgcn_wmma_f32_16x16x4_f32(false, a0, false, bs,  (short)0, acc_s0, false, false);
        acc_s1 = __builtin_amdgcn_wmma_f32_16x16x4_f32(false, a1, false, bs,  (short)0, acc_s1, false, false);
        acc_s2 = __builtin_amdgcn_wmma_f32_16x16x4_f32(false, a2, false, bs,  (short)0, acc_s2, false, false);
        acc_v0 = __builtin_amdgcn_wmma_f32_16x16x4_f32(false, av, false, bv0, (short)0, acc_v0, false, false);
        acc_v1 = __builtin_amdgcn_wmma_f32_16x16x4_f32(false, av, false, bv1, (short)0, acc_v1, false, false);
        acc_v2 = __builtin_amdgcn_wmma_f32_16x16x4_f32(false, av, false, bv2, (short)0, acc_v2, false, false);
    }

    // ---------- transpose through LDS (reuse staging space) ----------
    __syncthreads();
    const int OB = h * 1536;                 // per-head output staging: 768 s + 768 v
    #pragma unroll
    for (int rr = 0; rr < 8; ++rr) {
        const int mrow = 8 * half + rr;      // D-row this vgpr holds
        { int ch =  0 + mrow; lds[OB + pos * 48 + ch] = acc_s0[rr] + bs_g[h * 48 + ch]; }
        { int ch = 16 + mrow; lds[OB + pos * 48 + ch] = acc_s1[rr] + bs_g[h * 48 + ch]; }
        { int ch = 32 + mrow; lds[OB + pos * 48 + ch] = acc_s2[rr] + bs_g[h * 48 + ch]; }
        const int vb = OB + 768 + pos * 48 + mrow * 3;
        lds[vb + 0] = acc_v0[rr];
        lds[vb + 1] = acc_v1[rr];
        lds[vb + 2] = acc_v2[rr];
    }
    __syncthreads();

    // ---------- coalesced output: 768 contiguous f32 per head for each output ----------
    {
        float* so = sout_g + ((size_t)h * NPOS + p0) * 48;
        float* vo = vout_g + ((size_t)h * NPOS + p0) * 48;
        #pragma unroll
        for (int i = 0; i < 6; ++i) {
            int idx = lane + i * 32;                     // 0..191 float4s
            v4f a = *(const v4f*)&lds[OB + idx * 4];
            *(v4f*)&so[idx * 4] = a;
            v4f b = *(const v4f*)&lds[OB + 768 + idx * 4];
            *(v4f*)&vo[idx * 4] = b;
        }
    }
}

extern "C" void kernel_launch(void* const* d_in, const int* in_sizes, int n_in,
                              void* d_out, int out_size, void* d_ws, size_t ws_size,
                              hipStream_t stream) {
    (void)in_sizes; (void)n_in; (void)out_size; (void)d_ws; (void)ws_size;
    const float* s  = (const float*)d_in[0];
    const float* v  = (const float*)d_in[1];
    const float* r  = (const float*)d_in[2];
    const float* w  = (const float*)d_in[3];
    const float* ws = (const float*)d_in[4];
    const float* bs = (const float*)d_in[5];
    const float* wv = (const float*)d_in[6];
    float* sout = (float*)d_out;
    float* vout = sout + (size_t)NHEAD * NPOS * SOUT;

    dim3 grid(NPOS / TILE);   // 8192 blocks x 256 threads (8 waves = 8 heads)
    equi_wmma_kernel<<<grid, 256, 0, stream>>>(s, v, r, w, ws, bs, wv, sout, vout);
}